// PixelMamba_28750511080011
// MI455X (gfx1250) — compile-verified
//
#include <hip/hip_runtime.h>
#include <hip/hip_bf16.h>

// ---------------- model constants ----------------
constexpr int HH    = 160;
constexpr int WW    = 160;
constexpr int HWC   = HH * WW;        // 25600
constexpr int LSEQ  = 2 * HWC;        // 51200 tokens
constexpr int DM    = 128;            // d_model
constexpr int DI    = 512;            // d_inner
constexpr int DSTATE= 256;
constexpr int NH    = 16;
constexpr int HD    = 32;
constexpr int DCONV = 4;
constexpr int CHK   = 64;
constexpr int CONVD = DI + 2 * DSTATE;          // 1024
constexpr int DIP   = 2*DI + 2*DSTATE + NH;     // 1552
constexpr int NC    = LSEQ / CHK;               // 800 chunks
constexpr float EPSV = 1e-5f;

// ---------------- WMMA f32 16x16x4 ----------------
typedef __attribute__((ext_vector_type(2))) float v2f;
typedef __attribute__((ext_vector_type(8))) float v8f;

__device__ __forceinline__ v8f wmma4(v2f a, v2f b, v8f c) {
  // D = A(16x4) * B(4x16) + C(16x16), fp32 exact
  return __builtin_amdgcn_wmma_f32_16x16x4_f32(false, a, false, b, (short)0, c, false, false);
}

__device__ __forceinline__ float siluf(float v) { return v / (1.f + expf(-v)); }
__device__ __forceinline__ float softplusf(float t) { return (t > 20.f) ? t : log1pf(expf(t)); }

// ---------------- K1: embed  u = xc @ fc0_w^T + b ----------------
__global__ void k_embed(const float* __restrict__ x, const float* __restrict__ w,
                        const float* __restrict__ b, float* __restrict__ u) {
  int t = blockIdx.x * blockDim.x + threadIdx.x;        // LSEQ*DM threads
  int l = t >> 7, d = t & 127;
  int j = (l < HWC) ? l : (l - HWC);
  int h = j / WW, wpx = j % WW;
  if (l >= HWC) { h = HH - 1 - h; wpx = WW - 1 - wpx; }
  int p = h * WW + wpx;
  float acc = b[d];
  #pragma unroll
  for (int c = 0; c < 3; ++c) acc += x[c * HWC + p] * w[d * 3 + c];
  u[(size_t)l * DM + d] = acc;
}

// ---------------- K2a: in_proj GEMM, 16x64 per wave (z + xBC columns) ----------------
__global__ void __launch_bounds__(256)
k_inproj(const float* __restrict__ u, const float* __restrict__ wip,
         float* __restrict__ z, float* __restrict__ xbcr) {
  const int wave = blockIdx.x * (blockDim.x >> 5) + (threadIdx.x >> 5); // 3200*24 waves exactly
  const int lane = threadIdx.x & 31;
  const int mt = wave / 24, ng = wave % 24;          // n-group of 64 cols, covers 0..1535
  const int m0 = mt * 16, n0 = ng * 64;
  const int r = lane & 15, ko = (lane >> 4) * 2, hi = lane >> 4;
  v8f acc0 = {0.f}, acc1 = {0.f}, acc2 = {0.f}, acc3 = {0.f};
  const size_t rowA = (size_t)(m0 + r) * DM;
  const size_t rowB0 = (size_t)(n0      + r) * DM;   // B[k][n] = wip[n][k]
  const size_t rowB1 = (size_t)(n0 + 16 + r) * DM;
  const size_t rowB2 = (size_t)(n0 + 32 + r) * DM;
  const size_t rowB3 = (size_t)(n0 + 48 + r) * DM;
  for (int k0 = 0; k0 < DM; k0 += 4) {
    v2f a, b0, b1, b2, b3;
    a.x  = u[rowA + k0 + ko];      a.y  = u[rowA + k0 + ko + 1];
    b0.x = wip[rowB0 + k0 + ko];   b0.y = wip[rowB0 + k0 + ko + 1];
    b1.x = wip[rowB1 + k0 + ko];   b1.y = wip[rowB1 + k0 + ko + 1];
    b2.x = wip[rowB2 + k0 + ko];   b2.y = wip[rowB2 + k0 + ko + 1];
    b3.x = wip[rowB3 + k0 + ko];   b3.y = wip[rowB3 + k0 + ko + 1];
    acc0 = wmma4(a, b0, acc0);
    acc1 = wmma4(a, b1, acc1);
    acc2 = wmma4(a, b2, acc2);
    acc3 = wmma4(a, b3, acc3);
  }
  #pragma unroll
  for (int g = 0; g < 4; ++g) {
    v8f acc = (g == 0) ? acc0 : (g == 1) ? acc1 : (g == 2) ? acc2 : acc3;
    int j = n0 + g * 16 + r;                        // < 1536 always
    #pragma unroll
    for (int rr = 0; rr < 8; ++rr) {
      int m = m0 + rr + 8 * hi;
      float v = acc[rr];
      if (j < DI) z[(size_t)m * DI + j] = v;
      else        xbcr[(size_t)m * CONVD + (j - DI)] = v;
    }
  }
}

// ---------------- K2b: in_proj dt columns (1536..1551) + softplus ----------------
__global__ void __launch_bounds__(256)
k_inproj_dt(const float* __restrict__ u, const float* __restrict__ wip,
            const float* __restrict__ dt_bias, float* __restrict__ dt) {
  const int wave = blockIdx.x * (blockDim.x >> 5) + (threadIdx.x >> 5); // 3200 waves exactly
  const int lane = threadIdx.x & 31;
  const int m0 = wave * 16;
  const int n0 = DIP - NH;                            // 1536
  const int r = lane & 15, ko = (lane >> 4) * 2, hi = lane >> 4;
  v8f acc = {0.f};
  const size_t rowA = (size_t)(m0 + r) * DM;
  const size_t rowB = (size_t)(n0 + r) * DM;
  for (int k0 = 0; k0 < DM; k0 += 4) {
    v2f a, b;
    a.x = u[rowA + k0 + ko];   a.y = u[rowA + k0 + ko + 1];
    b.x = wip[rowB + k0 + ko]; b.y = wip[rowB + k0 + ko + 1];
    acc = wmma4(a, b, acc);
  }
  #pragma unroll
  for (int rr = 0; rr < 8; ++rr) {
    int m = m0 + rr + 8 * hi;
    dt[(size_t)m * NH + r] = softplusf(acc[rr] + dt_bias[r]);
  }
}

// ---------------- K3: causal depthwise conv (w=4) + bias + SiLU ----------------
__global__ void k_conv(const float* __restrict__ xbcr, const float* __restrict__ cw,
                       const float* __restrict__ cb, float* __restrict__ xbca) {
  size_t t = (size_t)blockIdx.x * blockDim.x + threadIdx.x;  // LSEQ*CONVD
  int c = (int)(t % CONVD);
  int l = (int)(t / CONVD);
  float v = cb[c];
  #pragma unroll
  for (int k = 0; k < DCONV; ++k) {
    int ll = l - (DCONV - 1) + k;
    if (ll >= 0) v += xbcr[(size_t)ll * CONVD + c] * cw[c * DCONV + k];
  }
  xbca[(size_t)l * CONVD + c] = siluf(v);
}

// ---------------- K4: per-chunk inclusive cumsum of dt*A ----------------
__global__ void k_acs(const float* __restrict__ dt, const float* __restrict__ A_log,
                      float* __restrict__ acs, float* __restrict__ atot) {
  int t = blockIdx.x * blockDim.x + threadIdx.x;   // NC*NH = 12800
  int h = t & (NH - 1), c = t >> 4;
  float A = -expf(A_log[h]);
  float run = 0.f;
  for (int q = 0; q < CHK; ++q) {
    size_t l = (size_t)c * CHK + q;
    run += dt[l * NH + h] * A;
    acs[l * NH + h] = run;
  }
  atot[c * NH + h] = expf(run);
}

// ---------------- K5: G = C @ B^T per chunk, 16x64 per wave ----------------
__global__ void __launch_bounds__(256)
k_gmat(const float* __restrict__ xbca, float* __restrict__ G) {
  const int wave = blockIdx.x * (blockDim.x >> 5) + (threadIdx.x >> 5); // 800*4 waves exactly
  const int lane = threadIdx.x & 31;
  const int c = wave >> 2, mt = wave & 3;
  const int m0 = mt * 16;                              // q-tile; s covers all 64
  const int r = lane & 15, ko = (lane >> 4) * 2, hi = lane >> 4;
  const size_t base = (size_t)c * CHK * CONVD;
  const size_t rowC  = base + (size_t)(m0 + r) * CONVD + (DI + DSTATE); // Cm row
  const size_t rowB0 = base + (size_t)(     r) * CONVD + DI;            // Bm rows
  const size_t rowB1 = base + (size_t)(16 + r) * CONVD + DI;
  const size_t rowB2 = base + (size_t)(32 + r) * CONVD + DI;
  const size_t rowB3 = base + (size_t)(48 + r) * CONVD + DI;
  v8f acc0 = {0.f}, acc1 = {0.f}, acc2 = {0.f}, acc3 = {0.f};
  for (int k0 = 0; k0 < DSTATE; k0 += 4) {
    v2f a, b0, b1, b2, b3;
    a.x  = xbca[rowC  + k0 + ko];  a.y  = xbca[rowC  + k0 + ko + 1];
    b0.x = xbca[rowB0 + k0 + ko];  b0.y = xbca[rowB0 + k0 + ko + 1];
    b1.x = xbca[rowB1 + k0 + ko];  b1.y = xbca[rowB1 + k0 + ko + 1];
    b2.x = xbca[rowB2 + k0 + ko];  b2.y = xbca[rowB2 + k0 + ko + 1];
    b3.x = xbca[rowB3 + k0 + ko];  b3.y = xbca[rowB3 + k0 + ko + 1];
    acc0 = wmma4(a, b0, acc0);
    acc1 = wmma4(a, b1, acc1);
    acc2 = wmma4(a, b2, acc2);
    acc3 = wmma4(a, b3, acc3);
  }
  const size_t gBase = (size_t)c * (CHK * CHK);
  #pragma unroll
  for (int g = 0; g < 4; ++g) {
    v8f acc = (g == 0) ? acc0 : (g == 1) ? acc1 : (g == 2) ? acc2 : acc3;
    #pragma unroll
    for (int rr = 0; rr < 8; ++rr)
      G[gBase + (size_t)(m0 + rr + 8 * hi) * CHK + g * 16 + r] = acc[rr];
  }
}

// ---------------- K6: states[c,h] = (x*dt*decay)^T @ B, 16x64 per wave ----------------
__global__ void __launch_bounds__(256)
k_states(const float* __restrict__ xbca, const float* __restrict__ dt,
         const float* __restrict__ acs, float* __restrict__ st) {
  const int wave = blockIdx.x * (blockDim.x >> 5) + (threadIdx.x >> 5); // 800*16*2*4 exactly
  const int lane = threadIdx.x & 31;
  const int ng = wave & 3;            // n-group of 64 (DSTATE/64 = 4)
  const int pt = (wave >> 2) & 1;     // p-tile (2)
  const int hc = wave >> 3;
  const int h = hc & (NH - 1), c = hc >> 4;
  const int m0 = pt * 16, n0 = ng * 64;
  const int r = lane & 15, ko = (lane >> 4) * 2, hi = lane >> 4;
  const float acsLast = acs[((size_t)c * CHK + CHK - 1) * NH + h];
  v8f acc0 = {0.f}, acc1 = {0.f}, acc2 = {0.f}, acc3 = {0.f};
  for (int k0 = 0; k0 < CHK; k0 += 4) {
    int q0 = k0 + ko;
    size_t l0 = (size_t)c * CHK + q0;
    float s0 = dt[l0 * NH + h]     * expf(acsLast - acs[l0 * NH + h]);
    float s1 = dt[(l0+1) * NH + h] * expf(acsLast - acs[(l0+1) * NH + h]);
    v2f a, b0, b1, b2, b3;
    a.x = xbca[l0 * CONVD + h * HD + m0 + r]     * s0;
    a.y = xbca[(l0+1) * CONVD + h * HD + m0 + r] * s1;
    const size_t c0 = l0 * CONVD + DI + n0 + r;
    const size_t c1 = (l0+1) * CONVD + DI + n0 + r;
    b0.x = xbca[c0];      b0.y = xbca[c1];
    b1.x = xbca[c0 + 16]; b1.y = xbca[c1 + 16];
    b2.x = xbca[c0 + 32]; b2.y = xbca[c1 + 32];
    b3.x = xbca[c0 + 48]; b3.y = xbca[c1 + 48];
    acc0 = wmma4(a, b0, acc0);
    acc1 = wmma4(a, b1, acc1);
    acc2 = wmma4(a, b2, acc2);
    acc3 = wmma4(a, b3, acc3);
  }
  const size_t sBase = (((size_t)c * NH + h) * HD + m0) * DSTATE;
  #pragma unroll
  for (int g = 0; g < 4; ++g) {
    v8f acc = (g == 0) ? acc0 : (g == 1) ? acc1 : (g == 2) ? acc2 : acc3;
    #pragma unroll
    for (int rr = 0; rr < 8; ++rr)
      st[sBase + (size_t)(rr + 8 * hi) * DSTATE + n0 + g * 16 + r] = acc[rr];
  }
}

// ---------------- K7: sequential chunk scan (exclusive prefix, in place) ----------------
__global__ void k_scan(float* __restrict__ st, const float* __restrict__ atot) {
  int tid = blockIdx.x * blockDim.x + threadIdx.x;  // NH*HD*DSTATE = 131072
  int h = tid >> 13;                                // /(HD*DSTATE)
  float carry = 0.f;
  for (int c = 0; c < NC; ++c) {
    size_t off = (size_t)c * (NH * HD * DSTATE) + tid;
    if (c + 1 < NC)
      __builtin_prefetch(&st[off + (size_t)(NH * HD * DSTATE)], 0, 0); // hide HBM latency of next slab
    float s = st[off];
    st[off] = carry;                                // states_in[c] = prefix before chunk c
    carry = atot[c * NH + h] * carry + s;
  }
}

// ---------------- K8: Y = (L∘G)@X + exp(Acs)*(C@S^T) + D*x, dual p-tiles ----------------
__global__ void __launch_bounds__(256)
k_y(const float* __restrict__ xbca, const float* __restrict__ dt,
    const float* __restrict__ acs, const float* __restrict__ G,
    const float* __restrict__ st, const float* __restrict__ Dv,
    float* __restrict__ Y) {
  const int wave = blockIdx.x * (blockDim.x >> 5) + (threadIdx.x >> 5); // 800*16*4 exactly
  const int lane = threadIdx.x & 31;
  const int qt = wave & 3;
  const int hc = wave >> 2;
  const int h = hc & (NH - 1), c = hc >> 4;
  const int m0 = qt * 16;                                  // q-tile (4); both p-tiles handled
  const int r = lane & 15, ko = (lane >> 4) * 2, hi = lane >> 4;
  const int qA = m0 + r;                                   // A-frag row
  const float acsq = acs[((size_t)c * CHK + qA) * NH + h];
  const size_t gRow = (size_t)c * (CHK * CHK) + (size_t)qA * CHK;

  // part 1: diagonal block  (K = 64 over s)
  v8f d0 = {0.f}, d1 = {0.f};
  for (int k0 = 0; k0 < CHK; k0 += 4) {
    int s0 = k0 + ko;
    size_t l0 = (size_t)c * CHK + s0;
    v2f a, b0, b1;
    a.x = (s0     <= qA) ? G[gRow + s0]     * expf(acsq - acs[l0 * NH + h])     : 0.f;
    a.y = (s0 + 1 <= qA) ? G[gRow + s0 + 1] * expf(acsq - acs[(l0+1) * NH + h]) : 0.f;
    float w0 = dt[l0 * NH + h], w1 = dt[(l0+1) * NH + h];
    const size_t x0 = l0 * CONVD + h * HD + r;
    const size_t x1 = (l0+1) * CONVD + h * HD + r;
    b0.x = xbca[x0]      * w0;  b0.y = xbca[x1]      * w1;
    b1.x = xbca[x0 + 16] * w0;  b1.y = xbca[x1 + 16] * w1;
    d0 = wmma4(a, b0, d0);
    d1 = wmma4(a, b1, d1);
  }

  // part 2: off-diagonal via carried states  (K = 256 over n)
  v8f o0 = {0.f}, o1 = {0.f};
  const size_t rowC  = ((size_t)c * CHK + qA) * CONVD + (DI + DSTATE);
  const size_t rowS0 = (((size_t)c * NH + h) * HD +      r) * DSTATE;
  const size_t rowS1 = (((size_t)c * NH + h) * HD + 16 + r) * DSTATE;
  for (int k0 = 0; k0 < DSTATE; k0 += 4) {
    v2f a, b0, b1;
    a.x  = xbca[rowC + k0 + ko];  a.y  = xbca[rowC + k0 + ko + 1];
    b0.x = st[rowS0 + k0 + ko];   b0.y = st[rowS0 + k0 + ko + 1];
    b1.x = st[rowS1 + k0 + ko];   b1.y = st[rowS1 + k0 + ko + 1];
    o0 = wmma4(a, b0, o0);
    o1 = wmma4(a, b1, o1);
  }

  const float Dh = Dv[h];
  #pragma unroll
  for (int rr = 0; rr < 8; ++rr) {
    int q = m0 + rr + 8 * hi;
    size_t l = (size_t)c * CHK + q;
    float eq = expf(acs[l * NH + h]);
    int p0 = r, p1 = 16 + r;
    Y[l * DI + h * HD + p0] = d0[rr] + eq * o0[rr] + Dh * xbca[l * CONVD + h * HD + p0];
    Y[l * DI + h * HD + p1] = d1[rr] + eq * o1[rr] + Dh * xbca[l * CONVD + h * HD + p1];
  }
}

// ---------------- K9: y = RMSNorm(Y * silu(z)) * norm_w  (in place) ----------------
__global__ void k_gatenorm(float* __restrict__ Y, const float* __restrict__ z,
                           const float* __restrict__ nw) {
  __shared__ float red[256];
  int l = blockIdx.x, t = threadIdx.x;
  size_t base = (size_t)l * DI;
  float z0 = z[base + t],        z1 = z[base + t + 256];
  float g0 = Y[base + t]       * siluf(z0);
  float g1 = Y[base + t + 256] * siluf(z1);
  red[t] = g0 * g0 + g1 * g1;
  __syncthreads();
  for (int s = 128; s > 0; s >>= 1) { if (t < s) red[t] += red[t + s]; __syncthreads(); }
  float rms = rsqrtf(red[0] / DI + EPSV);
  Y[base + t]       = g0 * rms * nw[t];
  Y[base + t + 256] = g1 * rms * nw[t + 256];
}

// ---------------- K10: out_proj (second half only), 16x64 per wave ----------------
__global__ void __launch_bounds__(256)
k_outproj(const float* __restrict__ y, const float* __restrict__ wout,
          float* __restrict__ outm) {
  const int wave = blockIdx.x * (blockDim.x >> 5) + (threadIdx.x >> 5); // 1600*2 waves exactly
  const int lane = threadIdx.x & 31;
  const int mt = wave >> 1, ng = wave & 1;
  const int m0 = mt * 16, n0 = ng * 64;
  const int r = lane & 15, ko = (lane >> 4) * 2, hi = lane >> 4;
  const size_t rowA  = (size_t)(HWC + m0 + r) * DI;
  const size_t rowB0 = (size_t)(n0      + r) * DI;
  const size_t rowB1 = (size_t)(n0 + 16 + r) * DI;
  const size_t rowB2 = (size_t)(n0 + 32 + r) * DI;
  const size_t rowB3 = (size_t)(n0 + 48 + r) * DI;
  v8f acc0 = {0.f}, acc1 = {0.f}, acc2 = {0.f}, acc3 = {0.f};
  for (int k0 = 0; k0 < DI; k0 += 4) {
    v2f a, b0, b1, b2, b3;
    a.x  = y[rowA + k0 + ko];      a.y  = y[rowA + k0 + ko + 1];
    b0.x = wout[rowB0 + k0 + ko];  b0.y = wout[rowB0 + k0 + ko + 1];
    b1.x = wout[rowB1 + k0 + ko];  b1.y = wout[rowB1 + k0 + ko + 1];
    b2.x = wout[rowB2 + k0 + ko];  b2.y = wout[rowB2 + k0 + ko + 1];
    b3.x = wout[rowB3 + k0 + ko];  b3.y = wout[rowB3 + k0 + ko + 1];
    acc0 = wmma4(a, b0, acc0);
    acc1 = wmma4(a, b1, acc1);
    acc2 = wmma4(a, b2, acc2);
    acc3 = wmma4(a, b3, acc3);
  }
  #pragma unroll
  for (int g = 0; g < 4; ++g) {
    v8f acc = (g == 0) ? acc0 : (g == 1) ? acc1 : (g == 2) ? acc2 : acc3;
    #pragma unroll
    for (int rr = 0; rr < 8; ++rr)
      outm[(size_t)(m0 + rr + 8 * hi) * DM + n0 + g * 16 + r] = acc[rr];
  }
}

// ---------------- K11: row mean over d ----------------
__global__ void k_halfmean(const float* __restrict__ outm, float* __restrict__ hv) {
  int j = blockIdx.x * blockDim.x + threadIdx.x;  // HWC
  float s = 0.f;
  #pragma unroll 4
  for (int d = 0; d < DM; ++d) s += outm[(size_t)j * DM + d];
  hv[j] = s * (1.f / DM);
}

// ---------------- K12: final fc1 ----------------
__global__ void k_fc1(const float* __restrict__ hv, const float* __restrict__ w,
                      const float* __restrict__ b, float* __restrict__ out) {
  __shared__ float red[256];
  int d = blockIdx.x, t = threadIdx.x;
  float s = 0.f;
  for (int j = t; j < HWC; j += 256) s += hv[j] * w[(size_t)d * HWC + j];
  red[t] = s;
  __syncthreads();
  for (int k = 128; k > 0; k >>= 1) { if (t < k) red[t] += red[t + k]; __syncthreads(); }
  if (t == 0) out[d] = red[0] + b[d];
}

// ---------------- launcher ----------------
extern "C" void kernel_launch(void* const* d_in, const int* in_sizes, int n_in,
                              void* d_out, int out_size, void* d_ws, size_t ws_size,
                              hipStream_t stream) {
  const float* x        = (const float*)d_in[0];
  const float* fc0_w    = (const float*)d_in[1];
  const float* fc0_b    = (const float*)d_in[2];
  const float* in_proj  = (const float*)d_in[3];
  const float* conv_w   = (const float*)d_in[4];
  const float* conv_b   = (const float*)d_in[5];
  const float* dt_bias  = (const float*)d_in[6];
  const float* A_log    = (const float*)d_in[7];
  const float* Dvec     = (const float*)d_in[8];
  const float* norm_w   = (const float*)d_in[9];
  const float* out_proj = (const float*)d_in[10];
  const float* fc1_w    = (const float*)d_in[11];
  const float* fc1_b    = (const float*)d_in[12];
  float* out = (float*)d_out;

  float* ws = (float*)d_ws;
  float* pU   = ws;                               // LSEQ*DM
  float* pZ   = pU   + (size_t)LSEQ * DM;         // LSEQ*DI
  float* pXR  = pZ   + (size_t)LSEQ * DI;         // LSEQ*CONVD (pre-conv xBC)
  float* pDT  = pXR  + (size_t)LSEQ * CONVD;      // LSEQ*NH (softplus'd dt)
  float* pXA  = pDT  + (size_t)LSEQ * NH;         // LSEQ*CONVD (post conv+silu)
  float* pACS = pXA  + (size_t)LSEQ * CONVD;      // LSEQ*NH
  float* pAT  = pACS + (size_t)LSEQ * NH;         // NC*NH
  float* pG   = pAT  + (size_t)NC * NH;           // NC*64*64
  float* pST  = pG   + (size_t)NC * CHK * CHK;    // NC*NH*HD*DSTATE
  float* pY   = pST  + (size_t)NC * NH * HD * DSTATE; // LSEQ*DI
  float* pOM  = pY   + (size_t)LSEQ * DI;         // HWC*DM
  float* pHF  = pOM  + (size_t)HWC * DM;          // HWC

  // K1: embed
  k_embed<<<(LSEQ * DM) / 256, 256, 0, stream>>>(x, fc0_w, fc0_b, pU);
  // K2a: in_proj main GEMM: 3200*24 = 76800 waves, 8 waves/block
  k_inproj<<<76800 / 8, 256, 0, stream>>>(pU, in_proj, pZ, pXR);
  // K2b: in_proj dt tile: 3200 waves
  k_inproj_dt<<<3200 / 8, 256, 0, stream>>>(pU, in_proj, dt_bias, pDT);
  // K3: conv + silu
  k_conv<<<((size_t)LSEQ * CONVD) / 256, 256, 0, stream>>>(pXR, conv_w, conv_b, pXA);
  // K4: per-chunk cumsum of dt*A
  k_acs<<<(NC * NH) / 256, 256, 0, stream>>>(pDT, A_log, pACS, pAT);
  // K5: G = C @ B^T : 800*4 = 3200 waves
  k_gmat<<<3200 / 8, 256, 0, stream>>>(pXA, pG);
  // K6: chunk states : 800*16*2*4 = 102400 waves
  k_states<<<102400 / 8, 256, 0, stream>>>(pXA, pDT, pACS, pST);
  // K7: sequential scan over chunks (with next-slab prefetch)
  k_scan<<<(NH * HD * DSTATE) / 256, 256, 0, stream>>>(pST, pAT);
  // K8: Y = diag + off + D*x : 800*16*4 = 51200 waves
  k_y<<<51200 / 8, 256, 0, stream>>>(pXA, pDT, pACS, pG, pST, Dvec, pY);
  // K9: gate + RMSNorm (in place on pY)
  k_gatenorm<<<LSEQ, 256, 0, stream>>>(pY, pZ, norm_w);
  // K10: out_proj (second half of sequence only): 1600*2 = 3200 waves
  k_outproj<<<3200 / 8, 256, 0, stream>>>(pY, out_proj, pOM);
  // K11: per-token mean over DM
  k_halfmean<<<HWC / 256, 256, 0, stream>>>(pOM, pHF);
  // K12: final fc1 -> 128 outputs
  k_fc1<<<DM, 256, 0, stream>>>(pHF, fc1_w, fc1_b, out);

  (void)in_sizes; (void)n_in; (void)out_size; (void)ws_size; (void)pHF;
}